// KernelXML_30245159698785
// MI455X (gfx1250) — compile-verified
//
#include <hip/hip_runtime.h>
#include <hip/hip_bf16.h>
#include <math.h>

// ---------------------------------------------------------------------------
// SLAY extreme-classification loss, MI455X (gfx1250, wave32, WMMA f32 16x16x4)
// ---------------------------------------------------------------------------
// Shapes: H=4 D=128 P=16 M=32 R=2  EMBED=512  VOCAB=100000 L=32768 B=512 S=64 K=5
// Feature dim F = R*H*P*M = 4096.  phi is separable: phi = poly (x) prf, so we
// never materialize phi; we compute poly[N,H,P], prf[N,R,H,M] with fp32 WMMA and
// fold Z / dot-products analytically.

typedef __attribute__((ext_vector_type(2))) float v2f;
typedef __attribute__((ext_vector_type(8))) float v8f;

#define NHEAD 4
#define DHEAD 128
#define PDIM  16
#define MDIM  32
#define RQ    2
#define EMBED 512
#define NLAB  32768
#define BB    512
#define SS    64
#define KK    5

// Gauss-Laguerre R=2 nodes/weights, divided by C = 2 + 1e-6 (matches reference)
#define CC        2.000001f
#define NODE0     0.5857864376269049f
#define NODE1     3.4142135623730951f
#define WGT0      0.8535533905932737f
#define WGT1      0.14644660940672624f
#define INV_SQRT_M 0.17677669529663687f   // 1/sqrt(32)

#define ZPART_BLOCKS 256

__device__ __forceinline__ v8f wmma4(v2f a, v2f b, v8f c) {
  // V_WMMA_F32_16X16X4_F32 : D[16x16] = A[16x4] * B[4x16] + C
  return __builtin_amdgcn_wmma_f32_16x16x4_f32(false, a, false, b, (short)0, c,
                                               false, false);
}

// ---------------------------------------------------------------------------
// Kernel 1: masked mean embedding  query[b, :] = sum_s table[idx[b,s]] * m / sum m
// ---------------------------------------------------------------------------
__global__ __launch_bounds__(256) void mean_emb_kernel(
    const int* __restrict__ indices, const float* __restrict__ mask,
    const float* __restrict__ table, float* __restrict__ query) {
  const int b = blockIdx.x, t = threadIdx.x;
  float acc0 = 0.f, acc1 = 0.f, msum = 0.f;
  for (int s = 0; s < SS; ++s) {
    const int   id = indices[b * SS + s];
    const float mk = mask[b * SS + s];
    msum += mk;
    const float* row = table + (size_t)id * EMBED;
    acc0 += row[t] * mk;
    acc1 += row[t + 256] * mk;
  }
  const float inv = 1.0f / fmaxf(msum, 1e-9f);
  query[(size_t)b * EMBED + t]       = acc0 * inv;
  query[(size_t)b * EMBED + t + 256] = acc1 * inv;
}

// ---------------------------------------------------------------------------
// Kernel 2: SLAY factor kernel. One block = 256 threads = 8 waves; wave w owns
// (r = w>>2, h = w&3).  Processes 16-row tiles.
//   MODE 0: rows contiguous [N,512]; store poly/prf factors.
//   MODE 1: rows are columns of classifier_kernel [512, 32768]; accumulate Z
//           partials (poly^T * prf per tile, WMMA, K = tile rows).
//   MODE 2: rows gathered by labels (columns of classifier); store factors.
// ---------------------------------------------------------------------------
template <int MODE>
__global__ __launch_bounds__(256) void slay_feat_kernel(
    const float* __restrict__ X,        // MODE0: [N,512]; MODE1/2: CK [512, NLAB]
    const int* __restrict__ labels,     // MODE2 only
    int N,
    const float* __restrict__ anchors,  // [16,128]
    const float* __restrict__ omega,    // [2,4,128,32]
    float* __restrict__ polyOut,        // [N,64]   (MODE 0/2)
    float* __restrict__ prfOut,         // [N,256]  (MODE 0/2)
    float* __restrict__ zpart)          // [gridDim.x, 4096] (MODE 1)
{
  __shared__ float sh_x[16][516];       // normalized tile, padded stride
  __shared__ float sh_anchor[16][129];  // anchors[p][k]
  __shared__ float sh_stage[8][768];    // per-wave: [0..255] poly[n][p], [256..767] prf[n][m]
  __shared__ float sh_inv[16][NHEAD];
  __shared__ int   sh_lbl[16];

  const int t    = threadIdx.x;
  const int lane = t & 31;
  const int wv   = t >> 5;
  const int r    = wv >> 2;
  const int h    = wv & 3;
  const int col16 = lane & 15;
  const int kboff = (lane >= 16) ? 2 : 0;
  const int rowhi = (lane >= 16) ? 8 : 0;

  for (int i = t; i < PDIM * DHEAD; i += 256)
    sh_anchor[i >> 7][i & 127] = anchors[i];

  // per-(r,h) PRF constants
  const float node  = (r == 0) ? NODE0 : NODE1;
  const float wgt   = (r == 0) ? WGT0 : WGT1;
  const float s_r   = node / CC;
  const float sq2s  = sqrtf(2.0f * s_r);
  const float prfsc = sqrtf(wgt / CC) * INV_SQRT_M;
  const float* om   = omega + (size_t)((r * NHEAD + h) * DHEAD) * MDIM;

  v8f zacc[2];
  for (int i = 0; i < 8; ++i) { zacc[0][i] = 0.f; zacc[1][i] = 0.f; }

  const int numTiles = N >> 4;
  for (int tile = blockIdx.x; tile < numTiles; tile += gridDim.x) {
    const int n0 = tile << 4;
    __syncthreads();                       // protect LDS reuse across iterations
    if (MODE == 2 && t < 16) {
      int l = labels[n0 + t];
      sh_lbl[t] = (l < 0) ? 0 : l;
    }
    __syncthreads();

    // ---- load 16x512 tile into LDS ------------------------------------
    if (MODE == 0) {
      for (int j = 0; j < 32; ++j) {
        const int flat = j * 256 + t;
        const int row = flat >> 9, e = flat & 511;
        sh_x[row][e] = X[(size_t)(n0 + row) * EMBED + e];
      }
    } else {
      for (int j = 0; j < 32; ++j) {
        const int flat = j * 256 + t;
        const int e = flat >> 4, noff = flat & 15;
        const int col = (MODE == 2) ? sh_lbl[noff] : (n0 + noff);
        sh_x[noff][e] = X[(size_t)e * NLAB + col];
      }
    }
    __syncthreads();

    // ---- per-(row,head) safe-normalize --------------------------------
    if (t < 64) {
      const int row = t >> 2, hh = t & 3;
      float ss = 0.f;
      for (int k = 0; k < DHEAD; ++k) {
        const float v = sh_x[row][hh * DHEAD + k];
        ss += v * v;
      }
      sh_inv[row][hh] = 1.0f / fmaxf(sqrtf(ss), 1e-6f);
    }
    __syncthreads();
    for (int j = 0; j < 32; ++j) {
      const int flat = j * 256 + t;
      const int row = flat >> 9, e = flat & 511;
      sh_x[row][e] *= sh_inv[row][e >> 7];
    }
    __syncthreads();

    // ---- poly projection: [16x128] x anchors^T[128x16] (WMMA f32 K=4) --
    v8f pc; for (int i = 0; i < 8; ++i) pc[i] = 0.f;
    for (int kk = 0; kk < 32; ++kk) {
      const int kb = (kk << 2) + kboff;
      v2f a; a.x = sh_x[col16][h * DHEAD + kb];
             a.y = sh_x[col16][h * DHEAD + kb + 1];
      v2f b; b.x = sh_anchor[col16][kb];
             b.y = sh_anchor[col16][kb + 1];
      pc = wmma4(a, b, pc);
    }
    // square, 1/sqrt(P) scale; stage (and/or store)
    float* polyS = &sh_stage[wv][0];
    for (int i = 0; i < 8; ++i) {
      const int n = i + rowhi;
      float v = pc[i];
      v = v * v * 0.25f;
      if (MODE == 1) polyS[n * 16 + col16] = v;
      else if (r == 0)
        polyOut[(size_t)(n0 + n) * 64 + h * PDIM + col16] = v;
    }

    // ---- PRF projection: [16x128] x omega[r,h][128x32] -----------------
    float* prfS = &sh_stage[wv][256];
    for (int nt = 0; nt < 2; ++nt) {
      v8f c; for (int i = 0; i < 8; ++i) c[i] = 0.f;
      const int m = nt * 16 + col16;
      for (int kk = 0; kk < 32; ++kk) {
        const int kb = (kk << 2) + kboff;
        v2f a; a.x = sh_x[col16][h * DHEAD + kb];
               a.y = sh_x[col16][h * DHEAD + kb + 1];
        v2f b; b.x = om[(size_t)kb * MDIM + m];
               b.y = om[(size_t)(kb + 1) * MDIM + m];
        c = wmma4(a, b, c);
      }
      for (int i = 0; i < 8; ++i) {
        const int n = i + rowhi;
        float xv = c[i] * sq2s - s_r;
        xv = fminf(fmaxf(xv, -10.0f), 10.0f);
        const float v = __expf(xv) * prfsc;
        if (MODE == 1) prfS[n * MDIM + m] = v;
        else prfOut[(size_t)(n0 + n) * 256 + (r * NHEAD + h) * MDIM + m] = v;
      }
    }

    // ---- Z accumulation: poly^T[16x16] * prf[16x32], K = 16 rows -------
    if (MODE == 1) {
      for (int nt = 0; nt < 2; ++nt) {
        for (int kk = 0; kk < 4; ++kk) {
          const int kb = (kk << 2) + kboff;  // kb = tile-row index (K dim)
          v2f a; a.x = polyS[kb * 16 + col16];          // A[p][n] = poly[n][p]
                 a.y = polyS[(kb + 1) * 16 + col16];
          v2f b; b.x = prfS[kb * MDIM + nt * 16 + col16];
                 b.y = prfS[(kb + 1) * MDIM + nt * 16 + col16];
          zacc[nt] = wmma4(a, b, zacc[nt]);
        }
      }
    }
  }

  if (MODE == 1) {
    float* zp = zpart + (size_t)blockIdx.x * 4096 + (r * NHEAD + h) * (PDIM * MDIM);
    for (int nt = 0; nt < 2; ++nt)
      for (int i = 0; i < 8; ++i) {
        const int p = i + rowhi;
        const int m = nt * 16 + col16;
        zp[p * MDIM + m] = zacc[nt][i];
      }
  }
}

// ---------------------------------------------------------------------------
// Kernel 3: deterministic Z partial reduction (fixed summation order)
// ---------------------------------------------------------------------------
__global__ __launch_bounds__(256) void reduce_z_kernel(
    const float* __restrict__ zpart, float* __restrict__ Z) {
  const int i = blockIdx.x * 256 + threadIdx.x;  // 16 blocks x 256 = 4096
  float s = 0.f;
  for (int blk = 0; blk < ZPART_BLOCKS; ++blk)
    s += zpart[(size_t)blk * 4096 + i];
  Z[i] = s;
}

// ---------------------------------------------------------------------------
// Kernel 4: per-example loss via separable dot products. One block per b.
// thread t <-> (r = t>>7, h = (t>>5)&3, m = t&31), 256 = R*H*M exactly.
// ---------------------------------------------------------------------------
__global__ __launch_bounds__(256) void slay_loss_kernel(
    const float* __restrict__ polyQ, const float* __restrict__ prfQ,
    const float* __restrict__ polyP, const float* __restrict__ prfP,
    const float* __restrict__ Z, const float* __restrict__ label_mask,
    float* __restrict__ lossB) {
  __shared__ float shPQ[64];
  __shared__ float shPP[64];
  __shared__ float red[256];
  __shared__ float shLogZ;
  const int b = blockIdx.x, t = threadIdx.x;
  const int rh = t >> 5, hh = rh & 3, m = t & 31;

  if (t < 64) shPQ[t] = polyQ[(size_t)b * 64 + t];
  __syncthreads();

  const float pq = prfQ[(size_t)b * 256 + t];

  // denom = phiQ . Z
  float inner = 0.f;
  for (int p = 0; p < PDIM; ++p)
    inner += shPQ[hh * PDIM + p] * Z[rh * (PDIM * MDIM) + p * MDIM + m];
  red[t] = pq * inner;
  __syncthreads();
  for (int s = 128; s > 0; s >>= 1) {
    if (t < s) red[t] += red[t + s];
    __syncthreads();
  }
  if (t == 0) shLogZ = __logf(red[0] + 1e-6f);
  __syncthreads();
  const float logZv = shLogZ;

  float lacc = 0.f;  // only thread 0 accumulates
  for (int k = 0; k < KK; ++k) {
    const int n = b * KK + k;
    __syncthreads();
    if (t < 64) shPP[t] = polyP[(size_t)n * 64 + t];
    __syncthreads();
    const float pp = prfP[(size_t)n * 256 + t];
    float in2 = 0.f;
    for (int p = 0; p < PDIM; ++p)
      in2 += shPQ[hh * PDIM + p] * shPP[hh * PDIM + p];
    red[t] = pq * pp * in2;
    __syncthreads();
    for (int s = 128; s > 0; s >>= 1) {
      if (t < s) red[t] += red[t + s];
      __syncthreads();
    }
    if (t == 0) {
      const float num = red[0] + 1e-6f;
      lacc += -(__logf(num) - logZv) * label_mask[n];
    }
  }
  if (t == 0) lossB[b] = lacc;
}

__global__ __launch_bounds__(256) void final_kernel(
    const float* __restrict__ lossB, float* __restrict__ out) {
  __shared__ float red[256];
  const int t = threadIdx.x;
  red[t] = lossB[t] + lossB[t + 256];
  __syncthreads();
  for (int s = 128; s > 0; s >>= 1) {
    if (t < s) red[t] += red[t + s];
    __syncthreads();
  }
  if (t == 0) out[0] = red[0] * (1.0f / (float)BB);
}

// ---------------------------------------------------------------------------
extern "C" void kernel_launch(void* const* d_in, const int* in_sizes, int n_in,
                              void* d_out, int out_size, void* d_ws, size_t ws_size,
                              hipStream_t stream) {
  (void)in_sizes; (void)n_in; (void)out_size; (void)ws_size;
  const int*   indices    = (const int*)d_in[0];
  const float* mask       = (const float*)d_in[1];
  const int*   labels     = (const int*)d_in[2];
  const float* label_mask = (const float*)d_in[3];
  const float* table      = (const float*)d_in[4];
  const float* ck         = (const float*)d_in[5];  // [512, 32768]
  const float* omega      = (const float*)d_in[6];  // [2,4,128,32]
  const float* anchors    = (const float*)d_in[7];  // [16,128]
  float* out = (float*)d_out;
  float* ws  = (float*)d_ws;

  // workspace layout (floats)
  float* q      = ws;                      // 512*512      = 262144
  float* polyQ  = q      + 262144;         // 512*64       = 32768
  float* prfQ   = polyQ  + 32768;          // 512*256      = 131072
  float* polyP  = prfQ   + 131072;         // 2560*64      = 163840
  float* prfP   = polyP  + 163840;         // 2560*256     = 655360
  float* zpart  = prfP   + 655360;         // 256*4096     = 1048576
  float* Zbuf   = zpart  + 1048576;        // 4096
  float* lossB  = Zbuf   + 4096;           // 512

  // 1) masked mean embedding -> query
  mean_emb_kernel<<<BB, 256, 0, stream>>>(indices, mask, table, q);

  // 2) query factors (N=512 -> 32 tiles)
  slay_feat_kernel<0><<<32, 256, 0, stream>>>(q, nullptr, BB, anchors, omega,
                                              polyQ, prfQ, nullptr);

  // 3) classifier sweep -> Z partials (N=32768 -> 2048 tiles over 256 blocks)
  slay_feat_kernel<1><<<ZPART_BLOCKS, 256, 0, stream>>>(ck, nullptr, NLAB,
                                                        anchors, omega,
                                                        nullptr, nullptr, zpart);

  // 4) positive-label factors (N = B*K = 2560 -> 160 tiles)
  slay_feat_kernel<2><<<160, 256, 0, stream>>>(ck, labels, BB * KK, anchors,
                                               omega, polyP, prfP, nullptr);

  // 5) deterministic Z reduction
  reduce_z_kernel<<<16, 256, 0, stream>>>(zpart, Zbuf);

  // 6) per-example loss, 7) final scalar
  slay_loss_kernel<<<BB, 256, 0, stream>>>(polyQ, prfQ, polyP, prfP, Zbuf,
                                           label_mask, lossB);
  final_kernel<<<1, 256, 0, stream>>>(lossB, out);
}